// MB_projection_73547019976715
// MI455X (gfx1250) — compile-verified
//
#include <hip/hip_runtime.h>

// Fixed fan-in sparse projection: y[b,o] = sum_k vals[o,k] * x[b, idx[o,k]]
// B=1024, IN=1024, OUT=40960, K=7. Store-bandwidth bound (~168 MB out,
// roofline ~7.3 us @ 23.3 TB/s).
// CDNA5 paths: global_load_async_to_lds_b128 (ASYNCcnt) to stage x rows in
// LDS; per-lane ds_load_b32 gathers + v_fmac.

#define BLOCK   256
#define BT      16          // b-rows staged in LDS per block (16*1024*4B = 64 KB)
#define IN_DIM  1024
#define OUT_DIM 40960
#define KNZ     7
#define OCHUNKS 16
#define OCHUNK  (OUT_DIM / OCHUNKS)   // 2560 outputs per block
#define OPASSES (OCHUNK / BLOCK)      // 10 passes of 256 lanes

// Explicit LDS (addrspace 3) pointer type. AS3 pointers are 32-bit; their
// value is the LDS byte offset — exactly the VDST operand the async-load
// instruction wants.
typedef __attribute__((address_space(3))) float lds_float;

__global__ __launch_bounds__(BLOCK)
void MB_projection_fanin_kernel(const float* __restrict__ x,
                                const float* __restrict__ vals,
                                const int*   __restrict__ idx,
                                float*       __restrict__ y)
{
    __shared__ float xs[BT * IN_DIM];   // 64 KB, LDS offset 0 (sole allocation)
    lds_float* lxs = (lds_float*)xs;

    const int tid = threadIdx.x;
    const int b0  = blockIdx.y * BT;

    // ---- Stage 16 rows of x (one contiguous 64 KB span) into LDS via the
    //      CDNA5 async global->LDS path (tracked by ASYNCcnt).
    //      CRITICAL: pass the LDS *pointer* (not an integer offset) so the
    //      shared array escapes into the asm — otherwise clang proves xs is
    //      never written and folds every gather to poison. ------------------
    {
        const float* gsrc = x + (size_t)b0 * IN_DIM;
        #pragma unroll
        for (int i = 0; i < (BT * IN_DIM) / (4 * BLOCK); ++i) {   // 16 iters
            const int e = (i * BLOCK + tid) * 4;   // float index, 16B step
            lds_float*   lptr  = lxs + e;          // 32-bit AS3 VGPR = LDS offset
            const float* gaddr = gsrc + e;
            asm volatile("global_load_async_to_lds_b128 %0, %1, off"
                         :: "v"(lptr), "v"(gaddr)
                         : "memory");
        }
        // Tie the LDS base into the wait too, then barrier.
        asm volatile("s_wait_asynccnt 0" :: "v"(lxs) : "memory");
    }
    __syncthreads();

    const int o_base = blockIdx.x * OCHUNK;

    for (int p = 0; p < OPASSES; ++p) {
        const int o = o_base + p * BLOCK + tid;

        // Fan-in indices/weights for this output column, kept in registers.
        int   ix[KNZ];
        float vl[KNZ];
        #pragma unroll
        for (int k = 0; k < KNZ; ++k) {
            ix[k] = idx [(size_t)o * KNZ + k];
            vl[k] = vals[(size_t)o * KNZ + k];
        }

        // 16 b-rows: same gather index per row -> one VGPR address per k,
        // ds_load_b32 immediate offset covers b*4096 (<= 61440, fits 16-bit).
        #pragma unroll
        for (int b = 0; b < BT; ++b) {
            const int boff = b * IN_DIM;
            float acc = 0.0f;
            #pragma unroll
            for (int k = 0; k < KNZ; ++k)
                acc = fmaf(vl[k], lxs[boff + ix[k]], acc);
            // lane -> consecutive o: contiguous 128B store per wave
            y[(size_t)(b0 + b) * OUT_DIM + o] = acc;
        }
    }
}

extern "C" void kernel_launch(void* const* d_in, const int* in_sizes, int n_in,
                              void* d_out, int out_size, void* d_ws, size_t ws_size,
                              hipStream_t stream)
{
    const float* x    = (const float*)d_in[0];
    const float* vals = (const float*)d_in[1];
    const int*   idx  = (const int*)  d_in[2];
    float*       y    = (float*)d_out;

    dim3 grid(OCHUNKS, 1024 / BT);   // (16, 64) = 1024 blocks
    MB_projection_fanin_kernel<<<grid, BLOCK, 0, stream>>>(x, vals, idx, y);

    (void)in_sizes; (void)n_in; (void)out_size; (void)d_ws; (void)ws_size;
}